// PointNetSetAbstraction_90632399880315
// MI455X (gfx1250) — compile-verified
//
#include <hip/hip_runtime.h>
#include <hip/hip_bf16.h>

typedef __attribute__((ext_vector_type(2))) float v2f;
typedef __attribute__((ext_vector_type(8))) float v8f;

#define B_   16
#define N_   4096
#define C_   64
#define M_   1024
#define K_   32
#define EPS_ 1e-5f
#define BIGF 3.402823e38f

// ---------------------------------------------------------------------------
// Kernel 1: new_xyz[b,m,:] = xyz[b, indices[m], :]
// ---------------------------------------------------------------------------
__global__ __launch_bounds__(256) void k_new_xyz(const float* __restrict__ xyz,
                                                 const int* __restrict__ indices,
                                                 float* __restrict__ new_xyz) {
  int t = blockIdx.x * 256 + threadIdx.x;
  if (t >= B_ * M_) return;
  int b = t >> 10, m = t & 1023;
  int n = indices[m];
  const float* s = xyz + ((size_t)b * N_ + n) * 3;
  float* d = new_xyz + (size_t)t * 3;
  d[0] = s[0]; d[1] = s[1]; d[2] = s[2];
}

// ---------------------------------------------------------------------------
// Kernel 2: per query, compute 4096 squared distances (reference formula
// |q|^2 + |p|^2 - 2 q.p) into LDS, then extract the 32 smallest by iterative
// wave-wide argmin (tie-break: lowest index, matching jax.lax.top_k).
// One wave32 per query, 2 waves per 64-thread block.
// ---------------------------------------------------------------------------
__global__ __launch_bounds__(64) void k_knn(const float* __restrict__ xyz,
                                            const float* __restrict__ new_xyz,
                                            int* __restrict__ knn) {
  __shared__ float dist[2][N_];
  const int wave = threadIdx.x >> 5;
  const int lane = threadIdx.x & 31;
  const int q = blockIdx.x * 2 + wave;     // global query id in [0, B*M)
  const int b = q >> 10;

  float* dl = dist[wave];
  const float qx = new_xyz[q * 3 + 0];
  const float qy = new_xyz[q * 3 + 1];
  const float qz = new_xyz[q * 3 + 2];
  const float qq = qx * qx + qy * qy + qz * qz;
  const float* px = xyz + (size_t)b * N_ * 3;

  for (int n = lane; n < N_; n += 32) {
    float x = px[n * 3 + 0], y = px[n * 3 + 1], z = px[n * 3 + 2];
    dl[n] = qq + (x * x + y * y + z * z) - 2.0f * (qx * x + qy * y + qz * z);
  }
  __syncthreads();

  int* out = knn + q * K_;
  for (int k = 0; k < K_; ++k) {
    float mv = BIGF;
    int   mi = 0x7fffffff;
    for (int n = lane; n < N_; n += 32) {
      float v = dl[n];
      if (v < mv) { mv = v; mi = n; }
    }
#pragma unroll
    for (int off = 16; off > 0; off >>= 1) {
      float ov = __shfl_xor(mv, off, 32);
      int   oi = __shfl_xor(mi, off, 32);
      if (ov < mv || (ov == mv && oi < mi)) { mv = ov; mi = oi; }
    }
    if (lane == 0) out[k] = mi;
    dl[mi] = BIGF;                 // uniform index, uniform value
    __syncthreads();
  }
}

// ---------------------------------------------------------------------------
// f32 WMMA GEMM with fused BatchNorm + ReLU epilogue.
//   Y[rowsOut x 32] = relu( (W[rowsOut x kW] * X[kPad x 32] - mu) * g/sqrt(v+eps) + beta )
// X is in LDS, row-major with stride 32; kPad is a multiple of 4 (padded rows
// of X are zero). Uses V_WMMA_F32_16X16X4_F32 tiles per the ISA layout:
//   A 16x4 : lanes 0-15 -> K=0,1 ; lanes 16-31 -> K=2,3 (row = lane&15)
//   B 4x16 : lanes 0-15 -> K=0,1 ; lanes 16-31 -> K=2,3 (col = lane&15)
//   C/D    : VGPR r -> row r (lanes 0-15) / row r+8 (lanes 16-31)
// ---------------------------------------------------------------------------
__device__ __forceinline__ void gemm_bn_relu(
    const float* __restrict__ Xb, int kPad,
    const float* __restrict__ W, int kW, int rowsOut,
    const float* __restrict__ g, const float* __restrict__ bt,
    const float* __restrict__ mu, const float* __restrict__ vr,
    float* __restrict__ Yb, int lane)
{
  const int half = lane >> 4;
  const int l16  = lane & 15;
  const int nKt  = kPad >> 2;

  for (int mt = 0; mt < (rowsOut >> 4); ++mt) {
    const int arow = mt * 16 + l16;
    for (int nt = 0; nt < 2; ++nt) {
      const int col = nt * 16 + l16;
      v8f acc = {};
      for (int kt = 0; kt < nKt; ++kt) {
        const int k0 = kt * 4 + half * 2;
        v2f a, bf;
        a.x  = (k0 + 0 < kW) ? W[arow * kW + k0 + 0] : 0.0f;
        a.y  = (k0 + 1 < kW) ? W[arow * kW + k0 + 1] : 0.0f;
        bf.x = Xb[(k0 + 0) * 32 + col];
        bf.y = Xb[(k0 + 1) * 32 + col];
        acc = __builtin_amdgcn_wmma_f32_16x16x4_f32(
            /*neg_a=*/false, a, /*neg_b=*/false, bf,
            /*c_mod=*/(short)0, acc, /*reuse_a=*/false, /*reuse_b=*/false);
      }
      const int rbase = mt * 16 + half * 8;
#pragma unroll
      for (int r = 0; r < 8; ++r) {
        const int row = rbase + r;
        const float s = g[row] / sqrtf(vr[row] + EPS_);
        const float y = (acc[r] - mu[row]) * s + bt[row];
        Yb[row * 32 + col] = fmaxf(y, 0.0f);
      }
    }
  }
}

// ---------------------------------------------------------------------------
// Kernel 3: per query (one wave): gather X (67x32 -> padded 68x32) into LDS,
// run 3 fused GEMM+BN+ReLU layers via f32 WMMA, max-pool over the 32
// neighbors into out[b, o, m].
// ---------------------------------------------------------------------------
__global__ __launch_bounds__(64) void k_mlp(
    const float* __restrict__ features, const float* __restrict__ xyz,
    const float* __restrict__ new_xyz, const int* __restrict__ knn,
    const float* __restrict__ W0, const float* __restrict__ g0,
    const float* __restrict__ b0, const float* __restrict__ m0,
    const float* __restrict__ v0,
    const float* __restrict__ W1, const float* __restrict__ g1,
    const float* __restrict__ b1, const float* __restrict__ m1,
    const float* __restrict__ v1,
    const float* __restrict__ W2, const float* __restrict__ g2,
    const float* __restrict__ b2, const float* __restrict__ m2,
    const float* __restrict__ v2,
    float* __restrict__ out_feat)
{
  __shared__ float Xs[2][68 * 32];
  __shared__ float Ys[2][128 * 32];
  const int wave = threadIdx.x >> 5;
  const int lane = threadIdx.x & 31;
  const int q = blockIdx.x * 2 + wave;
  const int b = q >> 10;
  const int m = q & 1023;

  float* Xb = Xs[wave];
  float* Yb = Ys[wave];

  // Gather: lane j owns neighbor column j.
  const int nj = knn[q * K_ + lane];
  const float* fb = features + (size_t)b * C_ * N_;
#pragma unroll 8
  for (int c = 0; c < C_; ++c)
    Xb[c * 32 + lane] = fb[(size_t)c * N_ + nj];
  const float* p  = xyz + ((size_t)b * N_ + nj) * 3;
  const float* qc = new_xyz + (size_t)q * 3;
  Xb[64 * 32 + lane] = p[0] - qc[0];
  Xb[65 * 32 + lane] = p[1] - qc[1];
  Xb[66 * 32 + lane] = p[2] - qc[2];
  Xb[67 * 32 + lane] = 0.0f;          // K-padding row for layer 0
  __syncthreads();

  gemm_bn_relu(Xb, 68, W0, 67,  64, g0, b0, m0, v0, Yb, lane);  // 64x32
  __syncthreads();
  gemm_bn_relu(Yb, 64, W1, 64,  64, g1, b1, m1, v1, Xb, lane);  // 64x32
  __syncthreads();
  gemm_bn_relu(Xb, 64, W2, 64, 128, g2, b2, m2, v2, Yb, lane);  // 128x32
  __syncthreads();

  // Max-pool over the 32 neighbor columns.
  for (int o = lane; o < 128; o += 32) {
    float mx = -BIGF;
#pragma unroll
    for (int j = 0; j < 32; ++j) mx = fmaxf(mx, Yb[o * 32 + j]);
    out_feat[((size_t)b * 128 + o) * M_ + m] = mx;
  }
}

// ---------------------------------------------------------------------------
extern "C" void kernel_launch(void* const* d_in, const int* in_sizes, int n_in,
                              void* d_out, int out_size, void* d_ws, size_t ws_size,
                              hipStream_t stream) {
  const float* xyz      = (const float*)d_in[0];
  const float* features = (const float*)d_in[1];
  const int*   indices  = (const int*)d_in[2];
  const float* W0 = (const float*)d_in[3];
  const float* g0 = (const float*)d_in[4];
  const float* b0 = (const float*)d_in[5];
  const float* m0 = (const float*)d_in[6];
  const float* v0 = (const float*)d_in[7];
  const float* W1 = (const float*)d_in[8];
  const float* g1 = (const float*)d_in[9];
  const float* b1 = (const float*)d_in[10];
  const float* m1 = (const float*)d_in[11];
  const float* v1 = (const float*)d_in[12];
  const float* W2 = (const float*)d_in[13];
  const float* g2 = (const float*)d_in[14];
  const float* b2 = (const float*)d_in[15];
  const float* m2 = (const float*)d_in[16];
  const float* v2 = (const float*)d_in[17];

  float* new_xyz  = (float*)d_out;                       // (16,1024,3)
  float* out_feat = new_xyz + (size_t)B_ * M_ * 3;       // (16,128,1024)
  int*   knn      = (int*)d_ws;                          // (16,1024,32) ints = 2 MB

  k_new_xyz<<<(B_ * M_ + 255) / 256, 256, 0, stream>>>(xyz, indices, new_xyz);
  k_knn<<<(B_ * M_) / 2, 64, 0, stream>>>(xyz, new_xyz, knn);
  k_mlp<<<(B_ * M_) / 2, 64, 0, stream>>>(features, xyz, new_xyz, knn,
      W0, g0, b0, m0, v0, W1, g1, b1, m1, v1, W2, g2, b2, m2, v2, out_feat);
}